// HAttention1D_86122684219920
// MI455X (gfx1250) — compile-verified
//
#include <hip/hip_runtime.h>
#include <hip/hip_bf16.h>

// HAttention1D for MI455X (gfx1250, wave32, WMMA f32 16x16x4).
// Workspace layout (floats): qp | kp | vp | yp  (each 32*8160*64) | ap (32*8160)
// Total ws requirement: 268,431,360 bytes.

typedef float v2f __attribute__((ext_vector_type(2)));
typedef float v8f __attribute__((ext_vector_type(8)));

#define BH       32
#define N0       8192
#define DIM      64
#define PYR_ROWS 8160   // sum_{l=1..8} 8192>>l

// ---------------------------------------------------------------------------
// 16x16 attention core (one wave):  S' = Apart(16x64) * Brows(16x64)^T
// A operand = partner K rows, B operand = Q rows (so C holds S'[n][m], i.e.
// per lane: column m = lane&15, rows n = r + 8*(lane>>4)).  After shfl_xor(16)
// each lane holds all 16 scores for its q-row m, already in WMMA-A layout
// for the second GEMM.
// ---------------------------------------------------------------------------
__device__ __forceinline__ void attn16(const float* __restrict__ arow,
                                       const float* __restrict__ brow,
                                       float qscale, int lane,
                                       float e[16], float& asum)
{
    const int m  = lane & 15;
    const int h2 = (lane >> 4) & 1;
    v8f acc;
#pragma unroll
    for (int i = 0; i < 8; ++i) acc[i] = 0.0f;
#pragma unroll
    for (int c = 0; c < 16; ++c) {
        const float* pa = arow + m * DIM + 4 * c + 2 * h2;
        const float* pb = brow + m * DIM + 4 * c + 2 * h2;
        v2f a; a.x = pa[0];          a.y = pa[1];
        v2f b; b.x = pb[0] * qscale; b.y = pb[1] * qscale;
        acc = __builtin_amdgcn_wmma_f32_16x16x4_f32(false, a, false, b,
                                                    (short)0, acc, false, false);
    }
#pragma unroll
    for (int r = 0; r < 8; ++r) {
        float o = __shfl_xor(acc[r], 16, 32);
        e[8 * h2 + r]       = acc[r];
        e[8 * (1 - h2) + r] = o;
    }
    float mx = e[0];
#pragma unroll
    for (int i = 1; i < 16; ++i) mx = fmaxf(mx, e[i]);
    asum = 0.0f;
#pragma unroll
    for (int i = 0; i < 16; ++i) { e[i] = expf(e[i] - mx); asum += e[i]; }
}

// One 16x16 output tile t of Y(16x64) = expS(16x16) * V(16x64)
__device__ __forceinline__ v8f yTile(const float e[16],
                                     const float* __restrict__ vrow,
                                     int t, int lane)
{
    const int m  = lane & 15;
    const int h2 = (lane >> 4) & 1;
    v8f y;
#pragma unroll
    for (int i = 0; i < 8; ++i) y[i] = 0.0f;
#pragma unroll
    for (int c = 0; c < 4; ++c) {
        v2f a; a.x = e[4 * c + 2 * h2]; a.y = e[4 * c + 2 * h2 + 1];
        const float* pv = vrow + (4 * c + 2 * h2) * DIM + 16 * t + m;
        v2f b; b.x = pv[0]; b.y = pv[DIM];
        y = __builtin_amdgcn_wmma_f32_16x16x4_f32(false, a, false, b,
                                                  (short)0, y, false, false);
    }
    return y;
}

// ---------------------------------------------------------------------------
// Kernel 1: build q/k/v pyramids, levels 1..8.  One WG per (bh, 256-row group).
// q/k mean-pool (q pre-scaled by 1/sqrt(64)=0.125 at level 1), v sum-pools.
// ---------------------------------------------------------------------------
__global__ __launch_bounds__(256) void hatt_pool(const float* __restrict__ q,
                                                 const float* __restrict__ k,
                                                 const float* __restrict__ v,
                                                 float* __restrict__ qp,
                                                 float* __restrict__ kp,
                                                 float* __restrict__ vp)
{
    __shared__ float bufA[128 * DIM];  // 32 KB
    __shared__ float bufB[64 * DIM];   // 16 KB
    constexpr int ROFF[9] = {0, 0, 4096, 6144, 7168, 7680, 7936, 8064, 8128};

    const int bh  = blockIdx.x >> 5;
    const int g   = blockIdx.x & 31;
    const int tid = threadIdx.x;

    const size_t srcBase = ((size_t)bh * N0 + (size_t)g * 256) * DIM;
    const size_t pyrBase = (size_t)bh * PYR_ROWS * DIM;

    const float* srcs[3] = {q + srcBase, k + srcBase, v + srcBase};
    float*       dsts[3] = {qp + pyrBase, kp + pyrBase, vp + pyrBase};
    const float  w1s[3]  = {0.0625f, 0.5f, 1.0f};   // q: 0.5*0.125
    const float  wls[3]  = {0.5f, 0.5f, 1.0f};

    for (int a = 0; a < 3; ++a) {
        const float* src = srcs[a];
        float*       pyr = dsts[a];
        const float  w1 = w1s[a], wl = wls[a];
        // level 1: 128 rows, straight from global
        for (int idx = tid; idx < 128 * DIM; idx += 256) {
            int j = idx >> 6, cc = idx & 63;
            float val = (src[(2 * j) * DIM + cc] + src[(2 * j + 1) * DIM + cc]) * w1;
            bufA[idx] = val;
            pyr[(size_t)(ROFF[1] + g * 128 + j) * DIM + cc] = val;
        }
        __syncthreads();
        float* cur = bufA;
        float* nxt = bufB;
        for (int l = 2; l <= 8; ++l) {
            const int rows = 256 >> l;
            for (int idx = tid; idx < rows * DIM; idx += 256) {
                int j = idx >> 6, cc = idx & 63;
                float val = (cur[(2 * j) * DIM + cc] + cur[(2 * j + 1) * DIM + cc]) * wl;
                nxt[idx] = val;
                pyr[(size_t)(ROFF[l] + g * rows + j) * DIM + cc] = val;
            }
            __syncthreads();
            float* t2 = cur; cur = nxt; nxt = t2;
        }
    }
}

// ---------------------------------------------------------------------------
// Kernel 2: coarse levels 1..8, one wave per (bh, level, block); partner=B^1.
// Writes y/a pyramids.
// ---------------------------------------------------------------------------
__global__ __launch_bounds__(256) void hatt_coarse(const float* __restrict__ qp,
                                                   const float* __restrict__ kp,
                                                   const float* __restrict__ vp,
                                                   float* __restrict__ yp,
                                                   float* __restrict__ ap)
{
    const int wave = blockIdx.x * (blockDim.x >> 5) + (threadIdx.x >> 5);
    if (wave >= BH * 510) return;                 // wave-uniform exit
    const int lane = threadIdx.x & 31;
    const int bh = wave / 510;
    int r = wave - bh * 510;
    int l = 1;
    while (r >= (512 >> l)) { r -= (512 >> l); ++l; }
    const int B = r, P = B ^ 1;
    constexpr int ROFF[9] = {0, 0, 4096, 6144, 7168, 7680, 7936, 8064, 8128};

    const size_t base = (size_t)bh * PYR_ROWS * DIM;
    const float* qrow = qp + base + (size_t)(ROFF[l] + B * 16) * DIM;
    const float* krow = kp + base + (size_t)(ROFF[l] + P * 16) * DIM;
    const float* vrow = vp + base + (size_t)(ROFF[l] + P * 16) * DIM;

    float e[16], asum;
    attn16(krow, qrow, 1.0f, lane, e, asum);      // qp already carries 1/sqrt(d)

    const int m = lane & 15, h2 = (lane >> 4) & 1;
    float* yrow = yp + base + (size_t)(ROFF[l] + B * 16) * DIM;
#pragma unroll
    for (int t = 0; t < 4; ++t) {
        v8f y = yTile(e, vrow, t, lane);
#pragma unroll
        for (int rr = 0; rr < 8; ++rr)
            yrow[(rr + 8 * h2) * DIM + 16 * t + m] = y[rr];
    }
    if (h2 == 0)
        ap[(size_t)bh * PYR_ROWS + ROFF[l] + B * 16 + m] = asum;
}

// ---------------------------------------------------------------------------
// Kernel 3: level 0 (self-attention) fused with hierarchical combine.
// ---------------------------------------------------------------------------
__global__ __launch_bounds__(256) void hatt_final(const float* __restrict__ q,
                                                  const float* __restrict__ k,
                                                  const float* __restrict__ v,
                                                  const float* __restrict__ yp,
                                                  const float* __restrict__ ap,
                                                  float* __restrict__ out)
{
    const int wave = blockIdx.x * (blockDim.x >> 5) + (threadIdx.x >> 5);
    if (wave >= BH * 512) return;                 // wave-uniform exit
    const int lane = threadIdx.x & 31;
    const int bh = wave >> 9;
    const int B  = wave & 511;
    constexpr int ROFF[9] = {0, 0, 4096, 6144, 7168, 7680, 7936, 8064, 8128};

    const size_t inbase = ((size_t)bh * N0 + (size_t)B * 16) * DIM;
    float e[16], asum;
    attn16(k + inbase, q + inbase, 0.125f, lane, e, asum);

    const int m = lane & 15, h2 = (lane >> 4) & 1;
    const size_t pb = (size_t)bh * PYR_ROWS;

    // denominator for row m (then broadcast per output row via shfl)
    const int im = B * 16 + m;
    float dc = asum;
#pragma unroll
    for (int l = 1; l <= 8; ++l)
        dc += ap[pb + ROFF[l] + (im >> l)];
    float den[8];
#pragma unroll
    for (int rr = 0; rr < 8; ++rr)
        den[rr] = __shfl(dc, rr + 8 * h2, 32) + 1e-8f;

    const float* ypb  = yp + pb * DIM;
    const float* vrow = v + inbase;
#pragma unroll
    for (int t = 0; t < 4; ++t) {
        v8f y = yTile(e, vrow, t, lane);
        const int col = 16 * t + m;
        // levels >= 4: one coarse row per 16-row block (i>>l == B>>(l-4))
        float cy = 0.0f;
#pragma unroll
        for (int l = 4; l <= 8; ++l)
            cy += ypb[(size_t)(ROFF[l] + (B >> (l - 4))) * DIM + col];
#pragma unroll
        for (int rr = 0; rr < 8; ++rr) {
            const int i = B * 16 + rr + 8 * h2;
            float val = y[rr] + cy;
#pragma unroll
            for (int l = 1; l <= 3; ++l)
                val += ypb[(size_t)(ROFF[l] + (i >> l)) * DIM + col];
            out[((size_t)bh * N0 + i) * DIM + col] = val / den[rr];
        }
    }
}

// ---------------------------------------------------------------------------
extern "C" void kernel_launch(void* const* d_in, const int* in_sizes, int n_in,
                              void* d_out, int out_size, void* d_ws, size_t ws_size,
                              hipStream_t stream)
{
    const float* q = (const float*)d_in[0];
    const float* k = (const float*)d_in[1];
    const float* v = (const float*)d_in[2];
    float* out = (float*)d_out;
    float* ws  = (float*)d_ws;

    const size_t PYR = (size_t)BH * PYR_ROWS * DIM;  // 16,711,680 floats
    float* qp = ws;
    float* kp = qp + PYR;
    float* vp = kp + PYR;
    float* yp = vp + PYR;
    float* ap = yp + PYR;                            // BH*PYR_ROWS floats

    hatt_pool<<<BH * 32, 256, 0, stream>>>(q, k, v, qp, kp, vp);
    hatt_coarse<<<(BH * 510) / 8, 256, 0, stream>>>(qp, kp, vp, yp, ap);
    hatt_final<<<(BH * 512) / 8, 256, 0, stream>>>(q, k, v, yp, ap, out);
}